// CausalLocalSGU_2010044694853
// MI455X (gfx1250) — compile-verified
//
#include <hip/hip_runtime.h>

// ---------------------------------------------------------------------------
// CausalLocalSGU for MI455X (gfx1250, wave32)
//   b=4, n=4096, dim=2048, HEADS=4, WINDOW=128, d_out=1024, d_head=256
// Memory-bound: ~200 MB effective DRAM traffic, 8.6 GFLOP -> fp32 everywhere,
// V_WMMA_F32_16X16X4_F32 (exact fp32; weights ~1e-6 are fp16-subnormal, so
// low precision buys nothing on a bandwidth-limited op).
// Kernel 1 : per-token LN stats only (mean, rstd) -> 128 KB workspace.
// Kernel 2 : stages mask-applied W and LN-applied gate slab in LDS (~266 KB
//            of the 320 KB WGP pool), 8 waves x (16m x 128d) WMMA tiles,
//            fused bias + residual-gate epilogue.
// ---------------------------------------------------------------------------

typedef float v2f __attribute__((ext_vector_type(2)));
typedef float v8f __attribute__((ext_vector_type(8)));

#define HEADS   4
#define WINDOW  128
#define TWOW    256          // 2*WINDOW
#define DIM     2048
#define DOUT    1024
#define DHEAD   256
#define DHALF   128          // d_head split in two per workgroup
#define SEQ     4096
#define NWIN    32           // SEQ / WINDOW
#define LN_EPS  1e-5f

// LDS strides (floats), bank-conflict-free for the WMMA fragment reads:
//   W stride 260: bank(m,k) = (4m + k) mod 64 -> distinct across 16 m-lanes,
//                 K and K+2 half-wave columns never collide.
//   G stride 136: bank advances 8 per gate row, so the two 16-lane halves
//                 (rows k and k+2) hit disjoint bank sets.
#define WSTR 260
#define GSTR 136

// ---------------------------------------------------------------------------
// Kernel 1: per-token LayerNorm statistics (mean, rstd) -> ws
// one wave32 per token; 8 tokens per 256-thread block
// ---------------------------------------------------------------------------
__global__ __launch_bounds__(256)
void sgu_ln_stats_kernel(const float* __restrict__ x,
                         float2* __restrict__ stats)      // [4*4096] (mean,rstd)
{
    const int tid   = threadIdx.x;
    const int lane  = tid & 31;
    const int token = blockIdx.x * 8 + (tid >> 5);

    const float4* gp = (const float4*)(x + (size_t)token * DIM + DOUT);
    float s = 0.f, ss = 0.f;
    #pragma unroll
    for (int i = 0; i < 8; ++i) {            // 128 consecutive channels / iter
        const float4 g = gp[i * 32 + lane];
        s  += g.x + g.y + g.z + g.w;
        ss += g.x * g.x + g.y * g.y + g.z * g.z + g.w * g.w;
    }
    #pragma unroll
    for (int off = 16; off > 0; off >>= 1) {
        s  += __shfl_down(s,  off, 32);
        ss += __shfl_down(ss, off, 32);
    }
    if (lane == 0) {
        const float mean = s * (1.0f / DOUT);
        const float var  = ss * (1.0f / DOUT) - mean * mean;   // biased (jnp.var)
        stats[token] = make_float2(mean, rsqrtf(var + LN_EPS));
    }
}

// ---------------------------------------------------------------------------
// Kernel 2: per (batch-window, head, d-half) masked GEMM + bias + res-gate
//   C[128 x 128] = Wmask[h][128 x 256] * Gcat[256 x 128]
// ---------------------------------------------------------------------------
__global__ __launch_bounds__(256)
void sgu_gemm_kernel(const float* __restrict__ x,
                     const float* __restrict__ weight,    // [4][128][256]
                     const float* __restrict__ bias,      // [4][128]
                     const float* __restrict__ ln_gamma,  // [1024]
                     const float* __restrict__ ln_beta,   // [1024]
                     const float2* __restrict__ stats,    // [4*4096]
                     float* __restrict__ out)             // [4][4096][1024]
{
    extern __shared__ float smem[];
    float* Wlds = smem;                      // [128][WSTR]
    float* Glds = smem + WINDOW * WSTR;      // [256][GSTR]

    const int bw    = blockIdx.x;            // 0..127
    const int b     = bw >> 5;
    const int w_idx = bw & 31;
    const int h     = blockIdx.y;            // 0..3
    const int dh    = blockIdx.z;            // 0..1
    const int tid   = threadIdx.x;
    const int cbase = h * DHEAD + dh * DHALF;      // gate/out channel base

    // ---- stage causal-masked weight slice: zero where n > m + WINDOW ------
    {
        const int c4 = tid & 63;             // constant per thread
        const int n0 = c4 * 4;
        const float4* wg = (const float4*)(weight + (size_t)h * WINDOW * TWOW) + c4;
        #pragma unroll
        for (int k = 0; k < 32; ++k) {       // 128 rows / 4 rows-per-pass
            const int m = (tid >> 6) + k * 4;
            float4 v = wg[m * 64];
            if (n0 + 0 > m + WINDOW) v.x = 0.f;
            if (n0 + 1 > m + WINDOW) v.y = 0.f;
            if (n0 + 2 > m + WINDOW) v.z = 0.f;
            if (n0 + 3 > m + WINDOW) v.w = 0.f;
            *(float4*)(Wlds + m * WSTR + n0) = v;
        }
    }
    // ---- stage gate slab with LayerNorm applied on the fly ----------------
    //   rows 0..127 = previous window (zero if first), 128..255 = current
    {
        const int c4 = tid & 31;             // constant per thread
        const int ch = cbase + c4 * 4;
        const float4 gm = *(const float4*)(ln_gamma + ch);
        const float4 bt = *(const float4*)(ln_beta + ch);
        #pragma unroll
        for (int k = 0; k < 32; ++k) {       // 256 rows / 8 rows-per-pass
            const int row = (tid >> 5) + k * 8;
            float4 v;
            if (row < WINDOW && w_idx == 0) {
                v = make_float4(0.f, 0.f, 0.f, 0.f);
            } else {
                const size_t t  = (size_t)b * SEQ + (w_idx - 1) * WINDOW + row;
                const float2 st = stats[t];                      // mean, rstd
                const float4 g  = *(const float4*)(x + t * DIM + DOUT + ch);
                v.x = (g.x - st.x) * st.y * gm.x + bt.x;
                v.y = (g.y - st.x) * st.y * gm.y + bt.y;
                v.z = (g.z - st.x) * st.y * gm.z + bt.z;
                v.w = (g.w - st.x) * st.y * gm.w + bt.w;
            }
            *(float4*)(Glds + row * GSTR + c4 * 4) = v;
        }
    }
    __syncthreads();

    const int wave = tid >> 5;               // m-tile row 0..7
    const int lane = tid & 31;
    const int half = lane >> 4;              // WMMA half-wave (K pair select)
    const int l16  = lane & 15;

    v8f acc[8];
    #pragma unroll
    for (int c = 0; c < 8; ++c) acc[c] = (v8f){0.f,0.f,0.f,0.f,0.f,0.f,0.f,0.f};

    // A fragment (32-bit A 16x4): lanes 0-15 hold K,K+1; lanes 16-31 K+2,K+3
    const float* Wrow = Wlds + (wave * 16 + l16) * WSTR + 2 * half;

    #pragma unroll 4
    for (int kk = 0; kk < TWOW; kk += 4) {
        v2f a;
        a.x = Wrow[kk + 0];
        a.y = Wrow[kk + 1];
        const float* g0 = Glds + (kk + 2 * half) * GSTR + l16;
        #pragma unroll
        for (int c = 0; c < 8; ++c) {
            v2f bf;
            bf.x = g0[c * 16];               // B row K=kk+2*half, N = c*16+l16
            bf.y = g0[GSTR + c * 16];        // B row K+1
            acc[c] = __builtin_amdgcn_wmma_f32_16x16x4_f32(
                false, a, false, bf, (short)0, acc[c], false, false);
        }
    }

    // ---- epilogue: + bias[h][m], * residual x[..., ch], store -------------
    const float* bh = bias + h * WINDOW;
    #pragma unroll
    for (int c = 0; c < 8; ++c) {
        const int ch = cbase + c * 16 + l16; // output channel (also res channel)
        #pragma unroll
        for (int j = 0; j < 8; ++j) {
            const int m     = wave * 16 + half * 8 + j;   // C/D: VGPR j -> row
            const int token = w_idx * WINDOW + m;
            const size_t t  = (size_t)b * SEQ + token;
            const float val = acc[c][j] + bh[m];
            out[t * DOUT + ch] = val * x[t * DIM + ch];
        }
    }
}

// ---------------------------------------------------------------------------
extern "C" void kernel_launch(void* const* d_in, const int* in_sizes, int n_in,
                              void* d_out, int out_size, void* d_ws, size_t ws_size,
                              hipStream_t stream) {
    const float* x        = (const float*)d_in[0];   // [4,4096,2048]
    const float* weight   = (const float*)d_in[1];   // [4,128,256]
    const float* bias     = (const float*)d_in[2];   // [4,128]
    const float* ln_gamma = (const float*)d_in[3];   // [1024]
    const float* ln_beta  = (const float*)d_in[4];   // [1024]
    float*       out      = (float*)d_out;           // [4,4096,1024]
    float2*      stats    = (float2*)d_ws;           // 128 KB scratch

    sgu_ln_stats_kernel<<<dim3(4 * SEQ / 8), dim3(256), 0, stream>>>(x, stats);

    const size_t shmem = (size_t)(WINDOW * WSTR + TWOW * GSTR) * sizeof(float); // ~266 KB
    sgu_gemm_kernel<<<dim3(4 * NWIN, HEADS, 2), dim3(256), shmem, stream>>>(
        x, weight, bias, ln_gamma, ln_beta, stats, out);
}